// UCShiftPairwiseLoss_37623913513748
// MI455X (gfx1250) — compile-verified
//
#include <hip/hip_runtime.h>
#include <math.h>

typedef float v2f __attribute__((ext_vector_type(2)));
typedef float v8f __attribute__((ext_vector_type(8)));

#define NROW 8192
#define DIM  128
#define SS   30.0f
#define SENT -1e30f

// ---------------- kernel 1: unit-normalize rows of feat into ws ----------------
__global__ void norm_rows_kernel(const float* __restrict__ feat,
                                 float* __restrict__ fn) {
    int lane = threadIdx.x & 31;
    int row  = blockIdx.x * 8 + (threadIdx.x >> 5);
    const float4 v = *(const float4*)(feat + row * DIM + lane * 4);
    float ss = v.x * v.x + v.y * v.y + v.z * v.z + v.w * v.w;
    #pragma unroll
    for (int off = 16; off >= 1; off >>= 1)
        ss += __shfl_xor(ss, off, 32);
    float inv = 1.0f / fmaxf(sqrtf(ss), 1e-12f);
    float4 o; o.x = v.x * inv; o.y = v.y * inv; o.z = v.z * inv; o.w = v.w * inv;
    *(float4*)(fn + row * DIM + lane * 4) = o;
}

// merge LSE state (m1,s1) <- merge((m1,s1),(m2,s2))
__device__ __forceinline__ void lse_merge(float& m1, float& s1, float m2, float s2) {
    float M = fmaxf(m1, m2);
    s1 = s1 * __expf(m1 - M) + s2 * __expf(m2 - M);
    m1 = M;
}

// acos via A&S 4.4.46 degree-7 minimax: acos(u) = sqrt(1-u)*P7(u), |err| <= 2e-8.
// Requires x in [-1, 1] (caller clamps). Well-conditioned near |x| = 1:
// 1-u is exact (Sterbenz) and P7(1) = sqrt(2), matching acos(1-e) ~ sqrt(2e).
__device__ __forceinline__ float fast_acos(float x) {
    float u = fabsf(x);
    float p = fmaf(u, -1.2624911e-3f,  6.6700901e-3f);
    p = fmaf(u, p, -1.70881256e-2f);
    p = fmaf(u, p,  3.0891881e-2f);
    p = fmaf(u, p, -5.01743046e-2f);
    p = fmaf(u, p,  8.89789874e-2f);
    p = fmaf(u, p, -2.145988016e-1f);
    p = fmaf(u, p,  1.5707963050f);
    float r = sqrtf(1.0f - u) * p;
    return (x >= 0.0f) ? r : (3.14159265358979f - r);
}

// ---------------- kernel 2: tiled WMMA gram + fused epilogue + online LSE ------
// grid: (NROW/64, 4), block: 128 threads (4 waves of 16 rows each)
// part layout: part[row*16 + cb*4 + {mP,sP,mN,sN}]
__global__ void __launch_bounds__(128, 1)
pairwise_lse_kernel(const float* __restrict__ fn,
                    const float* __restrict__ uc,
                    const int*   __restrict__ lab,
                    float* __restrict__ part) {
    __shared__ float lds[64 * 132];   // 64 columns x 128 K, padded stride 132

    const int tid  = threadIdx.x;
    const int lane = tid & 31;
    const int wave = tid >> 5;
    const int half = lane >> 4;       // 0: lanes 0-15, 1: lanes 16-31
    const int l16  = lane & 15;

    const int rBase    = blockIdx.x * 64 + wave * 16;
    const int colStart = blockIdx.y * (NROW / 4);
    const int colEnd   = colStart + (NROW / 4);

    // ---- A operands: 16 rows x 128 K resident in registers (64 VGPRs) ----
    // A 16x4 f32 layout: lane L holds M=L&15; VGPR0 = K = half*2, VGPR1 = K = half*2+1
    v2f A[32];
    {
        const float* arow = fn + (rBase + l16) * DIM + half * 2;
        #pragma unroll
        for (int kk = 0; kk < 32; ++kk)
            A[kk] = *(const v2f*)(arow + kk * 4);
    }

    // per-lane row metadata: slot r -> row rBase + half*8 + r
    const int rOff = rBase + half * 8;
    float ucR[8]; int labR[8];
    #pragma unroll
    for (int r = 0; r < 8; ++r) { ucR[r] = uc[rOff + r]; labR[r] = lab[rOff + r]; }

    // online logsumexp state per row slot
    float mP[8], sP[8], mN[8], sN[8];
    #pragma unroll
    for (int r = 0; r < 8; ++r) { mP[r] = SENT; sP[r] = 0.f; mN[r] = SENT; sN[r] = 0.f; }

    for (int c0 = colStart; c0 < colEnd; c0 += 64) {
        __syncthreads();
        // ---- stage 64 columns of fn into LDS, 4 float4s in flight at a time ----
        #pragma unroll
        for (int jb = 0; jb < 4; ++jb) {
            float4 t[4];
            #pragma unroll
            for (int j = 0; j < 4; ++j) {
                int i   = tid + (jb * 4 + j) * 128;   // 0..2047 float4 tiles
                int col = i >> 5;                     // 32 float4 per column
                int k4  = (i & 31) * 4;
                t[j] = *(const float4*)(fn + (c0 + col) * DIM + k4);
            }
            #pragma unroll
            for (int j = 0; j < 4; ++j) {
                int i   = tid + (jb * 4 + j) * 128;
                int col = i >> 5;
                int k4  = (i & 31) * 4;
                *(float4*)(&lds[col * 132 + k4]) = t[j];
            }
        }
        __syncthreads();

        // prefetch next chunk's columns into cache while we compute
        if (c0 + 64 < colEnd)
            __builtin_prefetch(fn + (c0 + 64) * DIM + tid * 64, 0, 3);

        #pragma clang loop unroll(disable)
        for (int sub = 0; sub < 4; ++sub) {
            const int colJ  = c0 + sub * 16 + l16;
            const float ucC = uc[colJ];
            const int  labC = lab[colJ];

            // ---- 16x16 gram tile via 32 x v_wmma_f32_16x16x4_f32 ----
            v8f C = {};
            const float* bp = &lds[(sub * 16 + l16) * 132 + half * 2];
            #pragma unroll
            for (int kk = 0; kk < 32; ++kk) {
                v2f B = *(const v2f*)(bp + kk * 4);
                C = __builtin_amdgcn_wmma_f32_16x16x4_f32(
                        false, A[kk], false, B, (short)0, C, false, false);
            }

            // ---- fused epilogue: -acos(g), uc correction, online LSE ----
            #pragma unroll
            for (int r = 0; r < 8; ++r) {
                float g  = fminf(fmaxf(C[r], -1.0f), 1.0f);
                float th = fast_acos(g);
                bool  eq = (labR[r] == labC);
                // v = S * ((eq ? theta : -theta) - (uc_i + uc_j))
                float v  = SS * ((eq ? th : -th) - (ucR[r] + ucC));

                float m = eq ? mP[r] : mN[r];
                float s = eq ? sP[r] : sN[r];
                float d  = v - m;
                bool  gt = d > 0.0f;
                float e  = __expf(gt ? -d : d);     // exp(-|d|): 1 exp / element
                float ns = gt ? fmaf(s, e, 1.0f) : (s + e);
                float nm = gt ? v : m;
                mP[r] = eq ? nm : mP[r];  sP[r] = eq ? ns : sP[r];
                mN[r] = eq ? mN[r] : nm;  sN[r] = eq ? sN[r] : ns;
            }
        }
    }

    // ---- merge across the 16 lanes of each half-wave ----
    #pragma unroll
    for (int r = 0; r < 8; ++r) {
        #pragma unroll
        for (int off = 1; off < 16; off <<= 1) {
            float m2 = __shfl_xor(mP[r], off, 32);
            float s2 = __shfl_xor(sP[r], off, 32);
            lse_merge(mP[r], sP[r], m2, s2);
            m2 = __shfl_xor(mN[r], off, 32);
            s2 = __shfl_xor(sN[r], off, 32);
            lse_merge(mN[r], sN[r], m2, s2);
        }
    }

    // lanes 0 and 16 write partial states for their 8 rows
    if (l16 == 0) {
        #pragma unroll
        for (int r = 0; r < 8; ++r) {
            float* p = part + (size_t)(rOff + r) * 16 + blockIdx.y * 4;
            p[0] = mP[r]; p[1] = sP[r]; p[2] = mN[r]; p[3] = sN[r];
        }
    }
}

// ---------------- kernel 3: merge column-block partials, softplus, reduce -----
__global__ void finalize_kernel(const float* __restrict__ part,
                                float* __restrict__ out) {
    __shared__ float red[256];
    int tid = threadIdx.x;
    float acc = 0.0f;
    for (int row = tid; row < NROW; row += 256) {
        float mp = SENT, sp = 0.f, mn = SENT, sn = 0.f;
        #pragma unroll
        for (int cb = 0; cb < 4; ++cb) {
            const float* p = part + (size_t)row * 16 + cb * 4;
            lse_merge(mp, sp, p[0], p[1]);
            lse_merge(mn, sn, p[2], p[3]);
        }
        float lp = mp + logf(sp);
        float ln = mn + logf(sn);
        float x  = lp + ln;
        // softplus(x) = max(x,0) + log1p(exp(-|x|))
        acc += fmaxf(x, 0.0f) + log1pf(expf(-fabsf(x)));
    }
    red[tid] = acc;
    __syncthreads();
    #pragma unroll
    for (int off = 128; off >= 1; off >>= 1) {
        if (tid < off) red[tid] += red[tid + off];
        __syncthreads();
    }
    if (tid == 0) out[0] = red[0] / ((float)NROW * SS);
}

extern "C" void kernel_launch(void* const* d_in, const int* in_sizes, int n_in,
                              void* d_out, int out_size, void* d_ws, size_t ws_size,
                              hipStream_t stream) {
    const float* feat = (const float*)d_in[0];   // [8192,128] f32
    const float* uc   = (const float*)d_in[1];   // [8192] f32
    const int*   lab  = (const int*)d_in[2];     // [8192] int32
    float* out = (float*)d_out;

    float* fn   = (float*)d_ws;                      // 8192*128 f32 = 4 MB
    float* part = fn + (size_t)NROW * DIM;           // 8192*16  f32 = 512 KB

    norm_rows_kernel<<<NROW / 8, 256, 0, stream>>>(feat, fn);

    dim3 grid(NROW / 64, 4);
    pairwise_lse_kernel<<<grid, 128, 0, stream>>>(fn, uc, lab, part);

    finalize_kernel<<<1, 256, 0, stream>>>(part, out);
}